// EdgeTensLinear_16398185136913
// MI455X (gfx1250) — compile-verified
//
#include <hip/hip_runtime.h>

typedef float v2f __attribute__((ext_vector_type(2)));
typedef float v8f __attribute__((ext_vector_type(8)));

#define DIM      256             // DIM_IN == DIM_OUT
#define SP       3               // spatial
#define EPW      16              // edges per workgroup -> 48 columns = 3 N-tiles
#define NCOL     (EPW * SP)      // 48
#define EDGE_DW  (DIM * SP)      // 768 dwords per edge
#define TILE_DW  (EPW * EDGE_DW) // 12288 dwords per block tile
#define ROW2     98              // LDS pair-row stride: 48 cols * 2 + pad 2
#define YSTR     771             // padded per-edge stride for Y stage (dwords)
#define THREADS  256             // 8 waves (wave32)

// One-shot pack: Wp[(k>>1)*512 + o*2 + (k&1)] = W[o][k]
// => an A-fragment (K-pair for one O column) is a single coalesced b64 load.
__global__ __launch_bounds__(256) void wt_pack_kernel(const float* __restrict__ W,
                                                      float* __restrict__ Wp) {
    int idx = blockIdx.x * 256 + threadIdx.x;   // 65536 elements
    int o = idx >> 8;
    int k = idx & 255;
    Wp[(k >> 1) * (DIM * 2) + o * 2 + (k & 1)] = W[o * DIM + k];
}

__global__ __launch_bounds__(256) void edge_wmma_kernel(const float* __restrict__ X,
                                                        const float* __restrict__ Wp,
                                                        float* __restrict__ Y,
                                                        int nEdges) {
    // X stage: 128 pair-rows * 98 = 12544 dw. Y stage reuse: 16*771 = 12336 dw.
    __shared__ float S[128 * ROW2];

    const int tid  = threadIdx.x;
    const int lane = tid & 31;
    const int wave = tid >> 5;
    const int lo   = lane & 15;     // N / M position within tile
    const int hi   = lane >> 4;     // selects K pair (A/B) or M+8 half (C/D)
    const long long eBase = (long long)blockIdx.x * EPW;

    const float* xg = X + eBase * EDGE_DW;
    float*       yg = Y + eBase * EDGE_DW;

    // ---- Phase 1: stage 16 edges (48KB, contiguous in global) into
    //      S[(k>>1)*ROW2 + n*2 + (k&1)], n = 3*eL + i  (K-pairs interleaved).
    {
        int g  = tid;               // flat dword index into the 48KB tile
        int eL = 0;
        int k  = tid / SP;          // tid < 768, so eL = 0 initially
        int i  = tid - k * SP;
        #pragma unroll 4
        for (int r = 0; r < TILE_DW / THREADS; ++r) {
            float v = (eBase + eL < nEdges) ? xg[g] : 0.0f;
            S[(k >> 1) * ROW2 + (eL * SP + i) * 2 + (k & 1)] = v;
            // advance flat index by 256 = 3*85 + 1 without divisions
            g += THREADS;
            k += 85; i += 1;
            if (i >= SP)  { i -= SP; k += 1; }
            if (k >= DIM) { k -= DIM; eL += 1; }
        }
    }
    __syncthreads();

    // ---- Phase 2: each wave computes rows [32*wave, 32*wave+32) x all 48 columns
    v8f acc[2][3] = {};

    const int oBase = wave * 32;
    const float* wp0 = Wp + (oBase + lo) * 2;        // O-tile 0
    const float* wp1 = Wp + (oBase + 16 + lo) * 2;   // O-tile 1

#pragma unroll 4
    for (int k0 = 0; k0 < DIM; k0 += 4) {
        int kp = (k0 >> 1) + hi;                     // K-pair row index
        // A fragments: {W[o][2kp], W[o][2kp+1]}  -- one b64 each
        v2f a0 = *(const v2f*)(wp0 + (long)kp * (DIM * 2));
        v2f a1 = *(const v2f*)(wp1 + (long)kp * (DIM * 2));
        // B fragments: {X[2kp][n], X[2kp+1][n]}   -- one ds b64 each
        const float* bs = &S[kp * ROW2 + lo * 2];
        v2f b0 = *(const v2f*)(bs);
        v2f b1 = *(const v2f*)(bs + 32);
        v2f b2 = *(const v2f*)(bs + 64);

        acc[0][0] = __builtin_amdgcn_wmma_f32_16x16x4_f32(false, a0, false, b0, (short)0, acc[0][0], false, false);
        acc[1][0] = __builtin_amdgcn_wmma_f32_16x16x4_f32(false, a1, false, b0, (short)0, acc[1][0], false, false);
        acc[0][1] = __builtin_amdgcn_wmma_f32_16x16x4_f32(false, a0, false, b1, (short)0, acc[0][1], false, false);
        acc[1][1] = __builtin_amdgcn_wmma_f32_16x16x4_f32(false, a1, false, b1, (short)0, acc[1][1], false, false);
        acc[0][2] = __builtin_amdgcn_wmma_f32_16x16x4_f32(false, a0, false, b2, (short)0, acc[0][2], false, false);
        acc[1][2] = __builtin_amdgcn_wmma_f32_16x16x4_f32(false, a1, false, b2, (short)0, acc[1][2], false, false);
    }
    __syncthreads();   // everyone done reading S before it becomes the Y stage

    // ---- Phase 3: scatter D tiles into S as [eL][O][i] (padded stride 771)
#pragma unroll
    for (int ob = 0; ob < 2; ++ob) {
#pragma unroll
        for (int nb = 0; nb < 3; ++nb) {
            int n  = nb * 16 + lo;
            int eL = n / SP;
            int i  = n - eL * SP;
            float* yb = &S[eL * YSTR + (oBase + ob * 16 + hi * 8) * SP + i];
#pragma unroll
            for (int r = 0; r < 8; ++r) yb[r * SP] = acc[ob][nb][r];
        }
    }
    __syncthreads();

    // ---- Phase 4: contiguous coalesced store of 48KB
    {
        int g   = tid;
        int eL  = 0;
        int rem = tid;
        #pragma unroll 4
        for (int r = 0; r < TILE_DW / THREADS; ++r) {
            if (eBase + eL < nEdges) yg[g] = S[eL * YSTR + rem];
            g += THREADS;
            rem += THREADS;
            if (rem >= EDGE_DW) { rem -= EDGE_DW; eL += 1; }
        }
    }
}

extern "C" void kernel_launch(void* const* d_in, const int* in_sizes, int n_in,
                              void* d_out, int out_size, void* d_ws, size_t ws_size,
                              hipStream_t stream) {
    const float* x = (const float*)d_in[0];   // [E, 256, 3]
    const float* W = (const float*)d_in[1];   // [256, 256]
    float* out = (float*)d_out;               // [E, 256, 3]
    float* Wp  = (float*)d_ws;                // 256 KB packed W

    int nEdges = in_sizes[0] / EDGE_DW;

    hipLaunchKernelGGL(wt_pack_kernel, dim3((DIM * DIM) / 256), dim3(256), 0, stream,
                       W, Wp);

    int nWG = (nEdges + EPW - 1) / EPW;       // 6250 for E=100000
    hipLaunchKernelGGL(edge_wmma_kernel, dim3(nWG), dim3(THREADS), 0, stream,
                       x, Wp, out, nEdges);
}